// TopLabelCalibrationPlot_5583457484862
// MI455X (gfx1250) — compile-verified
//
#include <hip/hip_runtime.h>

// TopLabelCalibrationPlot for MI455X (gfx1250).
// Bandwidth-bound streaming reduction: 512MB of probas @ 23.3TB/s ~= 22us floor.
// Data movement via Tensor Data Mover (tensor_load_to_lds), double-buffered in LDS,
// per-row argmax via ds_load_b128 + wave32 shfl_xor butterfly, LDS f32 atomics for bins.

#define TILE_ROWS 32
#define CCOLS     128   // reference fixes C = 128
#define MAXB      64    // max bins supported (reference uses 15)

typedef unsigned int v4u __attribute__((ext_vector_type(4)));
typedef int          v8i __attribute__((ext_vector_type(8)));
typedef int          v4i __attribute__((ext_vector_type(4)));

// Issue one TDM load: 2D tile of TILE_ROWS x CCOLS f32 from global -> LDS.
// Descriptor layout per CDNA5 ISA sec 8.3/8.4 (D# group0 128b, group1 256b).
__device__ __forceinline__ void tdm_load_tile(const float* gptr,
                                              unsigned lds_byte,
                                              int rows_valid) {
    unsigned long long ga = (unsigned long long)(uintptr_t)gptr;
    v4u g0;
    g0.x = 1u;                                        // count=1 (valid), gather off
    g0.y = lds_byte;                                  // lds_addr (bytes)
    g0.z = (unsigned)(ga & 0xFFFFFFFFu);              // global_addr[31:0]
    g0.w = (unsigned)((ga >> 32) & 0x01FFFFFFu)       // global_addr[56:32]
         | (2u << 30);                                // type = 2 ("image")
    v8i g1;
    g1[0] = 0x00020000;                               // data_size=2 -> 4 bytes
    g1[1] = (CCOLS & 0xFFFF) << 16;                   // tensor_dim0[15:0]
    g1[2] = ((CCOLS >> 16) & 0xFFFF)                  // tensor_dim0[31:16]
          | ((rows_valid & 0xFFFF) << 16);            // tensor_dim1[15:0] (OOB rows -> zero)
    g1[3] = ((rows_valid >> 16) & 0xFFFF)             // tensor_dim1[31:16]
          | (CCOLS << 16);                            // tile_dim0 = 128
    g1[4] = TILE_ROWS;                                // tile_dim1 = 32, tile_dim2 = 0
    g1[5] = CCOLS;                                    // tensor_dim0_stride[31:0] (elements)
    g1[6] = 0;                                        // stride0 hi / dim1_stride lo
    g1[7] = 0;
    v4i gz = {0, 0, 0, 0};
#if __clang_major__ >= 23
    v8i gz8 = {0, 0, 0, 0, 0, 0, 0, 0};
    __builtin_amdgcn_tensor_load_to_lds(g0, g1, gz, gz, gz8, 0);
#else
    __builtin_amdgcn_tensor_load_to_lds(g0, g1, gz, gz, 0);
#endif
}

__global__ __launch_bounds__(256)
void calib_main(const float* __restrict__ probas,
                const int* __restrict__ labels,
                const int* __restrict__ nbp,
                float* __restrict__ ws,
                int N, int numTiles) {
    __shared__ __align__(16) float tileBuf[2][TILE_ROWS * CCOLS];   // 2 x 16KB
    __shared__ float hist[3 * MAXB];

    const int tid  = threadIdx.x;
    const int wave = tid >> 5;
    const int lane = tid & 31;
    const int nb   = nbp[0];
    const float fnb   = (float)nb;
    const float delta = 1.0f / fnb;   // linspace step; edges are k*delta

    for (int i = tid; i < 3 * MAXB; i += blockDim.x) hist[i] = 0.0f;

    const int stride = gridDim.x;
    const int t0     = blockIdx.x;

    // Prologue: issue first tile into buffer 0.
    if (t0 < numTiles && wave == 0) {
        int rem = N - t0 * TILE_ROWS;
        if (rem > TILE_ROWS) rem = TILE_ROWS;
        tdm_load_tile(probas + (size_t)t0 * TILE_ROWS * CCOLS,
                      (unsigned)(uintptr_t)&tileBuf[0][0], rem);
    }
    __syncthreads();

    int buf = 0;
    for (int tile = t0; tile < numTiles; tile += stride) {
        // (1) everyone done with the buffer we're about to overwrite
        __syncthreads();
        if (wave == 0) {
            int next = tile + stride;
            if (next < numTiles) {
                int rem = N - next * TILE_ROWS;
                if (rem > TILE_ROWS) rem = TILE_ROWS;
                tdm_load_tile(probas + (size_t)next * TILE_ROWS * CCOLS,
                              (unsigned)(uintptr_t)&tileBuf[buf ^ 1][0], rem);
                __builtin_amdgcn_s_wait_tensorcnt(1);  // oldest (current tile) done
            } else {
                __builtin_amdgcn_s_wait_tensorcnt(0);  // drain
            }
        }
        // (2) current buffer is valid for all waves
        __syncthreads();

        const float* tb = &tileBuf[buf][0];
        for (int r = wave; r < TILE_ROWS; r += 8) {
            long long grow = (long long)tile * TILE_ROWS + r;
            if (grow >= N) continue;  // uniform within the wave

            // Whole 512B row in one ds_load_b128 per lane (lane L owns cols 4L..4L+3).
            const float4 v = *reinterpret_cast<const float4*>(tb + r * CCOLS + lane * 4);
            float c = v.x; int idx = lane * 4;
            if (v.y > c) { c = v.y; idx = lane * 4 + 1; }
            if (v.z > c) { c = v.z; idx = lane * 4 + 2; }
            if (v.w > c) { c = v.w; idx = lane * 4 + 3; }

            // wave32 butterfly argmax; tie -> lowest index (first occurrence).
            #pragma unroll
            for (int m = 16; m >= 1; m >>= 1) {
                float oc = __shfl_xor(c, m, 32);
                int   oi = __shfl_xor(idx, m, 32);
                if (oc > c || (oc == c && oi < idx)) { c = oc; idx = oi; }
            }

            if (lane == 0) {
                // strict-inequality bin: (k*delta < c) && (c < (k+1)*delta);
                // values exactly on an edge belong to no bin (matches reference).
                int k = (int)(c * fnb);
                if (k > nb - 1) k = nb - 1;
                int bin = -1;
                #pragma unroll
                for (int d = -1; d <= 1; ++d) {
                    int kk = k + d;
                    if (kk < 0 || kk >= nb) continue;
                    float lo = (float)kk * delta;
                    float hi = (kk + 1 == nb) ? 1.0f : (float)(kk + 1) * delta;
                    if (lo < c && c < hi) { bin = kk; break; }
                }
                if (bin >= 0) {
                    float correct = (labels[grow] == idx) ? 1.0f : 0.0f;
                    atomicAdd(&hist[bin], 1.0f);            // ds_add_f32
                    atomicAdd(&hist[MAXB + bin], c);
                    atomicAdd(&hist[2 * MAXB + bin], correct);
                }
            }
        }
        buf ^= 1;
    }

    __syncthreads();
    for (int i = tid; i < 3 * MAXB; i += blockDim.x) {
        float v = hist[i];
        if (v != 0.0f) atomicAdd(&ws[i], v);   // global_atomic_add_f32, 15 bins hot
    }
}

__global__ void calib_init(float* __restrict__ ws) {
    int i = blockIdx.x * blockDim.x + threadIdx.x;
    if (i < 3 * MAXB) ws[i] = 0.0f;
}

__global__ void calib_fin(const float* __restrict__ ws,
                          const int* __restrict__ nbp,
                          float* __restrict__ out) {
    int i = threadIdx.x;
    int nb = nbp[0];
    if (i < nb) {
        float n  = ws[i];
        float cs = ws[MAXB + i];
        float as = ws[2 * MAXB + i];
        float nanv = __uint_as_float(0x7FC00000u);
        bool empty = (n == 0.0f);
        out[i]          = empty ? nanv : cs / n;   // confs
        out[nb + i]     = empty ? nanv : as / n;   // accs
        out[2 * nb + i] = n;                       // n_samples
    }
}

extern "C" void kernel_launch(void* const* d_in, const int* in_sizes, int n_in,
                              void* d_out, int out_size, void* d_ws, size_t ws_size,
                              hipStream_t stream) {
    const float* probas = (const float*)d_in[0];
    const int*   labels = (const int*)d_in[1];
    const int*   nbp    = (const int*)d_in[2];
    float* out = (float*)d_out;
    float* ws  = (float*)d_ws;

    const int N = in_sizes[1];                       // rows; C = in_sizes[0]/N = 128
    const int numTiles = (N + TILE_ROWS - 1) / TILE_ROWS;
    int grid = numTiles < 4096 ? numTiles : 4096;
    if (grid < 1) grid = 1;

    hipLaunchKernelGGL(calib_init, dim3(1), dim3(3 * MAXB), 0, stream, ws);
    hipLaunchKernelGGL(calib_main, dim3(grid), dim3(256), 0, stream,
                       probas, labels, nbp, ws, N, numTiles);
    hipLaunchKernelGGL(calib_fin, dim3(1), dim3(MAXB), 0, stream, ws, nbp, out);
}